// TAGBlock_33904471835034
// MI455X (gfx1250) — compile-verified
//
#include <hip/hip_runtime.h>

#define N_NODES  100000
#define N_EDGES  1600000
#define F_DIM    64
#define N_GRAPHS 64
#define K_HOPS   3
#define EPS_GN   1e-5f

#define W_STRIDE 66   // 64 + 2 pad floats: rows 2 apart land on different LDS banks

typedef __attribute__((ext_vector_type(2))) float v2f;
typedef __attribute__((ext_vector_type(8))) float v8f;

// ---------------------------------------------------------------- utilities
__global__ void zero_f32_kernel(float* __restrict__ p, int n) {
  int i = blockIdx.x * blockDim.x + threadIdx.x;
  if (i < n) p[i] = 0.0f;
}

// ---------------------------------------------------------------- gcn_norm
__global__ void deg_kernel(const int* __restrict__ dst, const float* __restrict__ ew,
                           float* __restrict__ deg, int E) {
  int e = blockIdx.x * blockDim.x + threadIdx.x;
  if (e < E) atomicAdd(&deg[dst[e]], ew[e]);
}

__global__ void dis_kernel(const float* __restrict__ deg, float* __restrict__ dis, int n) {
  int i = blockIdx.x * blockDim.x + threadIdx.x;
  if (i < n) {
    float d = deg[i];
    dis[i] = (d > 0.0f) ? rsqrtf(d) : 0.0f;
  }
}

__global__ void norm_kernel(const int* __restrict__ src, const int* __restrict__ dst,
                            const float* __restrict__ ew, const float* __restrict__ dis,
                            float* __restrict__ normv, int E) {
  int e = blockIdx.x * blockDim.x + threadIdx.x;
  if (e < E) normv[e] = dis[src[e]] * ew[e] * dis[dst[e]];
}

// ------------------------------------------------- SpMM hop: hn[dst] += norm * h[src]
// one thread per (edge, 4-feature chunk): float4 gather + 4 float atomics
__global__ void spmm_kernel(const float* __restrict__ normv, const int* __restrict__ src,
                            const int* __restrict__ dst, const float* __restrict__ h,
                            float* __restrict__ hn, int E) {
  long gid = (long)blockIdx.x * blockDim.x + threadIdx.x;
  int e = (int)(gid >> 4);
  int q = (int)(gid & 15);
  if (e >= E) return;
  float w = normv[e];
  int s = src[e], d = dst[e];
  float4 v = ((const float4*)(h + (long)s * F_DIM))[q];
  float* o = hn + (long)d * F_DIM + q * 4;
  atomicAdd(o + 0, w * v.x);
  atomicAdd(o + 1, w * v.y);
  atomicAdd(o + 2, w * v.z);
  atomicAdd(o + 3, w * v.w);
}

// ------------------------------------------------- WMMA f32 GEMM: out(+)= H[N,64] @ W[64,64]
// wave32: each wave owns a 16-row strip, 4 column tiles of 16 -> 16x64 output.
// W staged once per block into LDS (shared by the 4 waves), padded stride kills
// the lanes0-15 vs lanes16-31 bank aliasing (rows 2 apart).
// A 16x4 f32: lane L: m=L&15, VGPR j holds K = (L>>4)*2 + j
// B 4x16 f32: lane L: n=L&15, VGPR j holds K-row (L>>4)*2 + j
// C/D 16x16 f32: VGPR r: lanes 0-15 -> M=r, lanes 16-31 -> M=r+8; N=lane&15
__global__ void __launch_bounds__(128)
gemm64_wmma_kernel(const float* __restrict__ H, const float* __restrict__ W,
                   float* __restrict__ out, int n_rows, int accumulate) {
  __shared__ float Ws[F_DIM * W_STRIDE];

  // cooperative stage: 4096 floats as 2048 float2 chunks over 128 threads
  {
    const float2* Wv = (const float2*)W;
    for (int i = threadIdx.x; i < (F_DIM * F_DIM) / 2; i += 128) {
      int row = i >> 5;          // 32 float2 per 64-float row
      int c   = i & 31;
      *(float2*)(Ws + row * W_STRIDE + c * 2) = Wv[i];
    }
  }
  __syncthreads();

  const int wave = threadIdx.x >> 5;
  const int lane = threadIdx.x & 31;
  const int row0 = (blockIdx.x * 4 + wave) * 16;
  if (row0 >= n_rows) return;          // whole-wave uniform: EXEC stays all-ones for WMMA
  const int m  = lane & 15;
  const int hi = lane >> 4;

  v8f c[4];
  const v8f zacc = {0.f,0.f,0.f,0.f,0.f,0.f,0.f,0.f};
#pragma unroll
  for (int t = 0; t < 4; ++t) {
    if (accumulate) {
#pragma unroll
      for (int r = 0; r < 8; ++r)
        c[t][r] = out[(long)(row0 + r + hi * 8) * F_DIM + t * 16 + m];
    } else {
      c[t] = zacc;
    }
  }

#pragma unroll
  for (int k0 = 0; k0 < F_DIM; k0 += 4) {
    v2f a;
    a.x = H[(long)(row0 + m) * F_DIM + k0 + hi * 2 + 0];
    a.y = H[(long)(row0 + m) * F_DIM + k0 + hi * 2 + 1];
#pragma unroll
    for (int t = 0; t < 4; ++t) {
      v2f b;
      b.x = Ws[(k0 + hi * 2 + 0) * W_STRIDE + t * 16 + m];
      b.y = Ws[(k0 + hi * 2 + 1) * W_STRIDE + t * 16 + m];
      c[t] = __builtin_amdgcn_wmma_f32_16x16x4_f32(
          /*neg_a=*/false, a, /*neg_b=*/false, b,
          /*c_mod=*/(short)0, c[t], /*reuse_a=*/false, /*reuse_b=*/false);
    }
  }

#pragma unroll
  for (int t = 0; t < 4; ++t)
#pragma unroll
    for (int r = 0; r < 8; ++r)
      out[(long)(row0 + r + hi * 8) * F_DIM + t * 16 + m] = c[t][r];
}

// ------------------------------------------------- GraphNorm + pooling
__global__ void bias_gsum_kernel(float* __restrict__ outb, const float* __restrict__ bvec,
                                 const int* __restrict__ batch, float* __restrict__ cnt,
                                 float* __restrict__ gsum) {
  long t = (long)blockIdx.x * blockDim.x + threadIdx.x;
  if (t >= (long)N_NODES * F_DIM) return;
  int f = (int)(t & (F_DIM - 1));
  int i = (int)(t >> 6);
  int g = batch[i];
  float v = outb[t] + bvec[f];
  outb[t] = v;
  atomicAdd(&gsum[g * F_DIM + f], v);
  if (f == 0) atomicAdd(&cnt[g], 1.0f);
}

__global__ void mean_kernel(const float* __restrict__ gsum, const float* __restrict__ cnt,
                            float* __restrict__ gmean) {
  int t = blockIdx.x * blockDim.x + threadIdx.x;
  if (t >= N_GRAPHS * F_DIM) return;
  float c = fmaxf(cnt[t >> 6], 1.0f);
  gmean[t] = gsum[t] / c;
}

__global__ void var_kernel(const float* __restrict__ outb, const int* __restrict__ batch,
                           const float* __restrict__ gmean, const float* __restrict__ gms,
                           float* __restrict__ gvar) {
  long t = (long)blockIdx.x * blockDim.x + threadIdx.x;
  if (t >= (long)N_NODES * F_DIM) return;
  int f = (int)(t & (F_DIM - 1));
  int g = batch[(int)(t >> 6)];
  float cen = outb[t] - gmean[g * F_DIM + f] * gms[f];
  atomicAdd(&gvar[g * F_DIM + f], cen * cen);
}

__global__ void invstd_kernel(const float* __restrict__ gvar, const float* __restrict__ cnt,
                              float* __restrict__ ginv) {
  int t = blockIdx.x * blockDim.x + threadIdx.x;
  if (t >= N_GRAPHS * F_DIM) return;
  float c = fmaxf(cnt[t >> 6], 1.0f);
  ginv[t] = rsqrtf(gvar[t] / c + EPS_GN);
}

__global__ void final_kernel(const float* __restrict__ outb, const float* __restrict__ x,
                             const int* __restrict__ batch, const float* __restrict__ gmean,
                             const float* __restrict__ gms, const float* __restrict__ ginv,
                             const float* __restrict__ gnw, const float* __restrict__ gnb,
                             float* __restrict__ hemb, float* __restrict__ psum,
                             unsigned* __restrict__ pmax) {
  long t = (long)blockIdx.x * blockDim.x + threadIdx.x;
  if (t >= (long)N_NODES * F_DIM) return;
  int f = (int)(t & (F_DIM - 1));
  int g = batch[(int)(t >> 6)];
  float cen = outb[t] - gmean[g * F_DIM + f] * gms[f];
  float hn  = gnw[f] * cen * ginv[g * F_DIM + f] + gnb[f];
  float he  = fmaxf(hn + x[t], 0.0f);     // residual + ReLU -> he >= 0
  hemb[t] = he;
  atomicAdd(&psum[g * F_DIM + f], he);
  atomicMax(&pmax[g * F_DIM + f], __float_as_uint(he));  // valid: non-negative floats
}

__global__ void pool_kernel(const float* __restrict__ psum, const unsigned* __restrict__ pmax,
                            const float* __restrict__ cnt, float* __restrict__ flat) {
  int t = blockIdx.x * blockDim.x + threadIdx.x;
  if (t >= N_GRAPHS * F_DIM) return;
  int g = t >> 6, f = t & (F_DIM - 1);
  float c = fmaxf(cnt[g], 1.0f);
  flat[g * (2 * F_DIM) + f]         = psum[t] / c;
  flat[g * (2 * F_DIM) + F_DIM + f] = __uint_as_float(pmax[t]);
}

// ---------------------------------------------------------------- driver
extern "C" void kernel_launch(void* const* d_in, const int* in_sizes, int n_in,
                              void* d_out, int out_size, void* d_ws, size_t ws_size,
                              hipStream_t stream) {
  (void)in_sizes; (void)n_in; (void)out_size; (void)ws_size;
  const float* x    = (const float*)d_in[0];
  const int*   eidx = (const int*)d_in[1];
  const int*   batch= (const int*)d_in[2];
  const float* ew   = (const float*)d_in[3];
  const float* Wp   = (const float*)d_in[4];
  const float* bv   = (const float*)d_in[5];
  const float* gnw  = (const float*)d_in[6];
  const float* gnb  = (const float*)d_in[7];
  const float* gms  = (const float*)d_in[8];

  const int  N = N_NODES, E = N_EDGES, F = F_DIM, G = N_GRAPHS;
  const long NF = (long)N * F;
  const int  GF = G * F;

  const int* src = eidx;
  const int* dst = eidx + E;

  float* ws = (float*)d_ws;
  float* deg   = ws;  ws += N;
  float* dis   = ws;  ws += N;
  float* normv = ws;  ws += E;
  float* hA    = ws;  ws += NF;
  float* hB    = ws;  ws += NF;
  float* outb  = ws;  ws += NF;
  float* cnt   = ws;                       // stats region: G + 6*GF floats, contiguous
  float* gsum  = cnt  + G;
  float* gmean = gsum + GF;
  float* gvar  = gmean + GF;
  float* ginv  = gvar + GF;
  float* psum  = ginv + GF;
  unsigned* pmax = (unsigned*)(psum + GF); // zero bits == 0.0f == ReLU floor

  float* hemb = (float*)d_out;
  float* flat = hemb + NF;

  auto cdiv = [](long a, long b) { return (int)((a + b - 1) / b); };

  zero_f32_kernel<<<cdiv(N, 256), 256, 0, stream>>>(deg, N);
  zero_f32_kernel<<<cdiv(G + 6 * GF, 256), 256, 0, stream>>>(cnt, G + 6 * GF);

  deg_kernel <<<cdiv(E, 256), 256, 0, stream>>>(dst, ew, deg, E);
  dis_kernel <<<cdiv(N, 256), 256, 0, stream>>>(deg, dis, N);
  norm_kernel<<<cdiv(E, 256), 256, 0, stream>>>(src, dst, ew, dis, normv, E);

  const int gemm_grid = cdiv(N, 64);   // 4 waves/block, 16 rows/wave
  gemm64_wmma_kernel<<<gemm_grid, 128, 0, stream>>>(x, Wp + 0 * F * F, outb, N, 0);

  const float* hsrc = x;
  float* bufs[2] = { hA, hB };
  for (int k = 1; k <= K_HOPS; ++k) {
    float* hdst = bufs[(k - 1) & 1];
    zero_f32_kernel<<<cdiv(NF, 256), 256, 0, stream>>>(hdst, (int)NF);
    spmm_kernel<<<cdiv((long)E * 16, 256), 256, 0, stream>>>(normv, src, dst, hsrc, hdst, E);
    gemm64_wmma_kernel<<<gemm_grid, 128, 0, stream>>>(hdst, Wp + k * F * F, outb, N, 1);
    hsrc = hdst;
  }

  bias_gsum_kernel<<<cdiv(NF, 256), 256, 0, stream>>>(outb, bv, batch, cnt, gsum);
  mean_kernel     <<<cdiv(GF, 256), 256, 0, stream>>>(gsum, cnt, gmean);
  var_kernel      <<<cdiv(NF, 256), 256, 0, stream>>>(outb, batch, gmean, gms, gvar);
  invstd_kernel   <<<cdiv(GF, 256), 256, 0, stream>>>(gvar, cnt, ginv);
  final_kernel    <<<cdiv(NF, 256), 256, 0, stream>>>(outb, x, batch, gmean, gms, ginv,
                                                      gnw, gnb, hemb, psum, pmax);
  pool_kernel     <<<cdiv(GF, 256), 256, 0, stream>>>(psum, pmax, cnt, flat);
}